// MyLayer_42133629174406
// MI455X (gfx1250) — compile-verified
//
#include <hip/hip_runtime.h>

// CDNA5 (gfx1250) wave32. Broadcast-add fused into the matrix pipe:
//   D = A(1s, 16x4) x B(10.5s, 4x16) + C  ==  C + 42.0 elementwise (exact in fp32)
typedef __attribute__((ext_vector_type(2))) float v2f;   // A/B operand: 16x4 f32 = 2 VGPRs/lane
typedef __attribute__((ext_vector_type(4))) float f32x4; // 128-bit load/store unit
typedef __attribute__((ext_vector_type(8))) float v8f;   // C/D operand: 16x16 f32 = 8 VGPRs/lane

__global__ __launch_bounds__(256) void add42_wmma_kernel(
    const float* __restrict__ in, float* __restrict__ out,
    int nTiles, long long n) {
  const int lane          = threadIdx.x & 31;          // wave32
  const int wavesPerBlock = blockDim.x >> 5;
  const int wave          = blockIdx.x * wavesPerBlock + (threadIdx.x >> 5);
  const int numWaves      = gridDim.x * wavesPerBlock;

  v2f a; a[0] = 1.0f;  a[1] = 1.0f;    // A = ones(16x4)
  v2f b; b[0] = 10.5f; b[1] = 10.5f;   // B = 10.5*ones(4x16); sum_k A*B = 4*10.5 = 42

  // One 256-float tile per wave per iteration; loop trip count is wave-uniform,
  // so EXEC is all-1s at every WMMA (ISA requirement).
  for (long long t = wave; t < nTiles; t += numWaves) {
    const long long base = t * 256;
    // Coalesced 128-bit loads: wave covers [base, base+256) contiguously.
    f32x4 lo = *(const f32x4*)(in + base + lane * 4);
    f32x4 hi = *(const f32x4*)(in + base + 128 + lane * 4);

    v8f c;
    c[0] = lo[0]; c[1] = lo[1]; c[2] = lo[2]; c[3] = lo[3];
    c[4] = hi[0]; c[5] = hi[1]; c[6] = hi[2]; c[7] = hi[3];

    // 8-arg form: (neg_a, A, neg_b, B, c_mod, C, reuse_a, reuse_b)
    c = __builtin_amdgcn_wmma_f32_16x16x4_f32(
        /*neg_a=*/false, a, /*neg_b=*/false, b,
        /*c_mod=*/(short)0, c, /*reuse_a=*/false, /*reuse_b=*/false);

    f32x4 olo, ohi;
    olo[0] = c[0]; olo[1] = c[1]; olo[2] = c[2]; olo[3] = c[3];
    ohi[0] = c[4]; ohi[1] = c[5]; ohi[2] = c[6]; ohi[3] = c[7];

    // Non-temporal stores: output is write-once, keep the 128 MiB input
    // resident in the 192 MB L2 across graph replays.
    __builtin_nontemporal_store(olo, (f32x4*)(out + base + lane * 4));
    __builtin_nontemporal_store(ohi, (f32x4*)(out + base + 128 + lane * 4));
  }

  // Tail (n % 256): predicated scalar path, no WMMA under partial EXEC.
  // (Zero-length for 8192*4096, but keeps the kernel generic.)
  const long long tailStart = (long long)nTiles * 256;
  const long long gtid    = (long long)blockIdx.x * blockDim.x + threadIdx.x;
  const long long tstride = (long long)gridDim.x * blockDim.x;
  for (long long i = tailStart + gtid; i < n; i += tstride)
    out[i] = in[i] + 42.0f;
}

extern "C" void kernel_launch(void* const* d_in, const int* in_sizes, int n_in,
                              void* d_out, int out_size, void* d_ws, size_t ws_size,
                              hipStream_t stream) {
  const float* x = (const float*)d_in[0];
  float* out     = (float*)d_out;
  const long long n = (long long)out_size;      // 8192*4096 = 33,554,432
  const int nTiles  = (int)(n / 256);           // 131,072 wave-tiles

  // 2048 blocks x 256 threads = 16,384 waves; 8 tiles per wave via grid-stride.
  // Plenty of waves to saturate HBM while keeping per-wave loop work for ILP.
  int blocks = 2048;
  if (blocks < 1) blocks = 1;
  add42_wmma_kernel<<<blocks, 256, 0, stream>>>(x, out, nTiles, n);
}